// GRCNN_60885456388892
// MI455X (gfx1250) — compile-verified
//
#include <hip/hip_runtime.h>

typedef __attribute__((ext_vector_type(2))) float v2f;
typedef __attribute__((ext_vector_type(8))) float v8f;

#define DDIM 512
#define TBM 256   // block tile M
#define TBN 128   // block tile N
#define TBK 16    // K chunk
#define LP 18     // padded LDS stride (floats)

// ---------------------------------------------------------------------------
// dinv[row] = 1 / (sum_j adj[row, j] + 1e-10), one wave32 per row
// ---------------------------------------------------------------------------
__global__ __launch_bounds__(256) void rowsum_inv_kernel(
    const float* __restrict__ adj, float* __restrict__ dinv, int nrows) {
  int gw = (blockIdx.x * 256 + threadIdx.x) >> 5;
  if (gw >= nrows) return;
  int lane = threadIdx.x & 31;
  const float* row = adj + (size_t)gw * DDIM;
  float s = 0.f;
#pragma unroll
  for (int j = lane * 4; j < DDIM; j += 128) {
    float4 v = *(const float4*)&row[j];
    s += v.x + v.y + v.z + v.w;
  }
#pragma unroll
  for (int off = 16; off > 0; off >>= 1) s += __shfl_xor(s, off, 32);
  if (lane == 0) dinv[gw] = 1.0f / (s + 1e-10f);
}

// ---------------------------------------------------------------------------
// out[b] = (dinv ? diag(dinv[b]) : I) * A[b] @ X[b]
// A,X,out: [B,512,512] row-major fp32.  256x128 block tile, 8 waves,
// each wave 64x64 = 4x4 WMMA accumulators.  Double-buffered LDS.
// B staged transposed [n][k] so fragments are single ds_load_b64.
// ---------------------------------------------------------------------------
__global__ __launch_bounds__(256) void msg_gemm_kernel(
    const float* __restrict__ A, const float* __restrict__ X,
    const float* __restrict__ dinv, float* __restrict__ out) {
  __shared__ float As[2][TBM * LP];  // [m][k]
  __shared__ float Bs[2][TBN * LP];  // [n][k] (transposed)

  const int b = blockIdx.z;
  const int m0 = blockIdx.y * TBM, n0 = blockIdx.x * TBN;
  const size_t bofs = (size_t)b * DDIM * DDIM;
  const float* Ab = A + bofs;
  const float* Xb = X + bofs;
  const float* dv = dinv ? (dinv + (size_t)b * DDIM) : nullptr;

  const int tid = threadIdx.x;
  const int lane = tid & 31, wave = tid >> 5;
  const int wm0 = (wave >> 1) * 64, wn0 = (wave & 1) * 64;
  const int l16 = lane & 15, lh = lane >> 4;

  v8f zero8 = {0.f, 0.f, 0.f, 0.f, 0.f, 0.f, 0.f, 0.f};
  v8f acc[4][4];
#pragma unroll
  for (int mt = 0; mt < 4; ++mt)
#pragma unroll
    for (int nt = 0; nt < 4; ++nt) acc[mt][nt] = zero8;

  float4 ra[4], rb[2];

  // Register-stage chunk k0 from global (A scaled by dinv on the fly).
  auto gload = [&](int k0) {
#pragma unroll
    for (int i = 0; i < 4; ++i) {               // A: 256x16
      int idx = tid + i * 256;
      int r = idx >> 2, c = (idx & 3) * 4;
      float4 v = *(const float4*)&Ab[(size_t)(m0 + r) * DDIM + k0 + c];
      float s = dv ? dv[m0 + r] : 1.0f;
      v.x *= s; v.y *= s; v.z *= s; v.w *= s;
      ra[i] = v;
    }
#pragma unroll
    for (int i = 0; i < 2; ++i) {               // B: 16x128
      int idx = tid + i * 256;
      int r = idx >> 5, c = (idx & 31) * 4;
      rb[i] = *(const float4*)&Xb[(size_t)(k0 + r) * DDIM + n0 + c];
    }
  };
  auto lstore = [&](int buf) {
#pragma unroll
    for (int i = 0; i < 4; ++i) {
      int idx = tid + i * 256;
      int r = idx >> 2, c = (idx & 3) * 4;
      As[buf][r * LP + c + 0] = ra[i].x;
      As[buf][r * LP + c + 1] = ra[i].y;
      As[buf][r * LP + c + 2] = ra[i].z;
      As[buf][r * LP + c + 3] = ra[i].w;
    }
#pragma unroll
    for (int i = 0; i < 2; ++i) {               // transpose into [n][k]
      int idx = tid + i * 256;
      int r = idx >> 5, c = (idx & 31) * 4;     // r = k, c = n
      Bs[buf][(c + 0) * LP + r] = rb[i].x;
      Bs[buf][(c + 1) * LP + r] = rb[i].y;
      Bs[buf][(c + 2) * LP + r] = rb[i].z;
      Bs[buf][(c + 3) * LP + r] = rb[i].w;
    }
  };

  const int nch = DDIM / TBK;  // 32
  gload(0);
  lstore(0);
  for (int c = 0; c < nch; ++c) {
    const int cur = c & 1;
    if (c + 1 < nch) gload((c + 1) * TBK);
    if (c + 2 < nch) {  // deep prefetch hint for chunk c+2
      __builtin_prefetch(&Ab[(size_t)(m0 + (tid >> 2)) * DDIM + (c + 2) * TBK], 0, 1);
      __builtin_prefetch(&Xb[(size_t)((c + 2) * TBK + (tid >> 5)) * DDIM + n0 + (tid & 31) * 4], 0, 1);
    }
    __syncthreads();
#pragma unroll
    for (int kk = 0; kk < TBK; kk += 4) {
      const int kf = kk + lh * 2;
      v2f a[4], bf[4];
#pragma unroll
      for (int mt = 0; mt < 4; ++mt)
        a[mt] = *(const v2f*)&As[cur][(wm0 + mt * 16 + l16) * LP + kf];
#pragma unroll
      for (int nt = 0; nt < 4; ++nt)
        bf[nt] = *(const v2f*)&Bs[cur][(wn0 + nt * 16 + l16) * LP + kf];
#pragma unroll
      for (int mt = 0; mt < 4; ++mt)
#pragma unroll
        for (int nt = 0; nt < 4; ++nt)
          acc[mt][nt] = __builtin_amdgcn_wmma_f32_16x16x4_f32(
              false, a[mt], false, bf[nt], (short)0, acc[mt][nt], false, false);
    }
    if (c + 1 < nch) lstore(1 - cur);
  }

  float* ob = out + bofs;
#pragma unroll
  for (int mt = 0; mt < 4; ++mt)
#pragma unroll
    for (int i = 0; i < 8; ++i) {
      int m = m0 + wm0 + mt * 16 + i + 8 * lh;
#pragma unroll
      for (int nt = 0; nt < 4; ++nt) {
        int n = n0 + wn0 + nt * 16 + l16;
        ob[(size_t)m * DDIM + n] = acc[mt][nt][i];
      }
    }
}

// ---------------------------------------------------------------------------
// out[b,m,o] = relu( base[b,m,o] + mask[b,m] *
//                    sum_{si<nSrc} ( src_si[b,m,:] . W_si[o,:] + bias_si[o] ) )
// W_si: [512,512] row-major == already-transposed B operand.
// ---------------------------------------------------------------------------
__global__ __launch_bounds__(256) void lin_gemm_kernel(
    const float* s0, const float* s1, const float* s2, const float* s3,
    const float* w0, const float* w1, const float* w2, const float* w3,
    const float* bb0, const float* bb1, const float* bb2, const float* bb3,
    int nSrc, const float* __restrict__ base, const float* __restrict__ mask,
    float* __restrict__ out) {
  __shared__ float As[2][TBM * LP];  // [m][k]
  __shared__ float Ws[2][TBN * LP];  // [o][k]

  const int b = blockIdx.z;
  const int m0 = blockIdx.y * TBM, n0 = blockIdx.x * TBN;
  const size_t bofs = (size_t)b * DDIM * DDIM;

  const int tid = threadIdx.x;
  const int lane = tid & 31, wave = tid >> 5;
  const int wm0 = (wave >> 1) * 64, wn0 = (wave & 1) * 64;
  const int l16 = lane & 15, lh = lane >> 4;

  v8f zero8 = {0.f, 0.f, 0.f, 0.f, 0.f, 0.f, 0.f, 0.f};
  v8f acc[4][4];
#pragma unroll
  for (int mt = 0; mt < 4; ++mt)
#pragma unroll
    for (int nt = 0; nt < 4; ++nt) acc[mt][nt] = zero8;

  float4 ra[4], rb[2];

  // chunk c: source index si = c>>5, k0 = (c&31)*16
  auto gload = [&](int c) {
    const int si = c >> 5, k0 = (c & 31) * TBK;
    const float* Sp = (si == 0) ? s0 : (si == 1) ? s1 : (si == 2) ? s2 : s3;
    const float* Wp = (si == 0) ? w0 : (si == 1) ? w1 : (si == 2) ? w2 : w3;
    const float* Abp = Sp + bofs;
#pragma unroll
    for (int i = 0; i < 4; ++i) {               // A: 256x16
      int idx = tid + i * 256;
      int r = idx >> 2, cc = (idx & 3) * 4;
      ra[i] = *(const float4*)&Abp[(size_t)(m0 + r) * DDIM + k0 + cc];
    }
#pragma unroll
    for (int i = 0; i < 2; ++i) {               // W: 128 rows x 16 k
      int idx = tid + i * 256;
      int r = idx >> 2, cc = (idx & 3) * 4;     // r = o_local
      rb[i] = *(const float4*)&Wp[(size_t)(n0 + r) * DDIM + k0 + cc];
    }
  };
  auto lstore = [&](int buf) {
#pragma unroll
    for (int i = 0; i < 4; ++i) {
      int idx = tid + i * 256;
      int r = idx >> 2, cc = (idx & 3) * 4;
      As[buf][r * LP + cc + 0] = ra[i].x;
      As[buf][r * LP + cc + 1] = ra[i].y;
      As[buf][r * LP + cc + 2] = ra[i].z;
      As[buf][r * LP + cc + 3] = ra[i].w;
    }
#pragma unroll
    for (int i = 0; i < 2; ++i) {               // direct copy: already [o][k]
      int idx = tid + i * 256;
      int r = idx >> 2, cc = (idx & 3) * 4;
      Ws[buf][r * LP + cc + 0] = rb[i].x;
      Ws[buf][r * LP + cc + 1] = rb[i].y;
      Ws[buf][r * LP + cc + 2] = rb[i].z;
      Ws[buf][r * LP + cc + 3] = rb[i].w;
    }
  };

  const int nch = nSrc * (DDIM / TBK);
  gload(0);
  lstore(0);
  for (int c = 0; c < nch; ++c) {
    const int cur = c & 1;
    if (c + 1 < nch) gload(c + 1);
    __syncthreads();
#pragma unroll
    for (int kk = 0; kk < TBK; kk += 4) {
      const int kf = kk + lh * 2;
      v2f a[4], bf[4];
#pragma unroll
      for (int mt = 0; mt < 4; ++mt)
        a[mt] = *(const v2f*)&As[cur][(wm0 + mt * 16 + l16) * LP + kf];
#pragma unroll
      for (int nt = 0; nt < 4; ++nt)
        bf[nt] = *(const v2f*)&Ws[cur][(wn0 + nt * 16 + l16) * LP + kf];
#pragma unroll
      for (int mt = 0; mt < 4; ++mt)
#pragma unroll
        for (int nt = 0; nt < 4; ++nt)
          acc[mt][nt] = __builtin_amdgcn_wmma_f32_16x16x4_f32(
              false, a[mt], false, bf[nt], (short)0, acc[mt][nt], false,
              false);
    }
    if (c + 1 < nch) lstore(1 - cur);
  }

  // Epilogue: bias sum, mask, residual, relu.
  float bsum[4];
#pragma unroll
  for (int nt = 0; nt < 4; ++nt) {
    int o = n0 + wn0 + nt * 16 + l16;
    float s = bb0[o];
    if (nSrc > 1) s += bb1[o];
    if (nSrc > 2) s += bb2[o];
    if (nSrc > 3) s += bb3[o];
    bsum[nt] = s;
  }
  const float* baseb = base + bofs;
  float* ob = out + bofs;
#pragma unroll
  for (int mt = 0; mt < 4; ++mt)
#pragma unroll
    for (int i = 0; i < 8; ++i) {
      int m = m0 + wm0 + mt * 16 + i + 8 * lh;
      float mk = mask[(size_t)b * DDIM + m];
#pragma unroll
      for (int nt = 0; nt < 4; ++nt) {
        int n = n0 + wn0 + nt * 16 + l16;
        float v = baseb[(size_t)m * DDIM + n] + mk * (acc[mt][nt][i] + bsum[nt]);
        ob[(size_t)m * DDIM + n] = fmaxf(v, 0.f);
      }
    }
}

// ---------------------------------------------------------------------------
extern "C" void kernel_launch(void* const* d_in, const int* in_sizes, int n_in,
                              void* d_out, int out_size, void* d_ws,
                              size_t ws_size, hipStream_t stream) {
  (void)in_sizes; (void)n_in; (void)out_size; (void)ws_size;
  const float* node = (const float*)d_in[0];
  const float* rela = (const float*)d_in[1];
  const float* am   = (const float*)d_in[2];
  const float* rm   = (const float*)d_in[3];
  const float* adj1 = (const float*)d_in[4];
  const float* adj2 = (const float*)d_in[5];
  const float* adj3 = (const float*)d_in[6];
  const float* rsub = (const float*)d_in[7];
  const float* robj = (const float*)d_in[8];
  const float* n2r  = (const float*)d_in[9];
  const float* Wnn  = (const float*)d_in[10];
  const float* bnn  = (const float*)d_in[11];
  const float* Wnr  = (const float*)d_in[12];
  const float* bnr  = (const float*)d_in[13];
  const float* Wr   = (const float*)d_in[14];
  const float* br   = (const float*)d_in[15];

  const size_t BNDsz = (size_t)64 * 512 * 512;   // elements per [B,N,D] tensor
  const size_t DD = (size_t)512 * 512;           // elements per weight matrix
  float* X0 = (float*)d_ws;
  float* X1 = X0 + BNDsz;
  float* X2 = X1 + BNDsz;
  float* X3 = X2 + BNDsz;
  float* X4 = X3 + BNDsz;  // node1
  float* X5 = X4 + BNDsz;  // rela1
  float* dv1 = X5 + BNDsz;
  float* dv2 = dv1 + (size_t)64 * 512;
  float* dv3 = dv2 + (size_t)64 * 512;
  float* outN = (float*)d_out;
  float* outR = outN + BNDsz;

  dim3 blk(256);
  dim3 gG(DDIM / TBN, DDIM / TBM, 64);  // (4, 2, 64)
  const int nrows = 64 * 512;

  // Row-sum inverses for the three adjacency matrices (fixed across steps).
  rowsum_inv_kernel<<<4096, blk, 0, stream>>>(adj1, dv1, nrows);
  rowsum_inv_kernel<<<4096, blk, 0, stream>>>(adj2, dv2, nrows);
  rowsum_inv_kernel<<<4096, blk, 0, stream>>>(adj3, dv3, nrows);

  // -------- step 1 : node --------
  msg_gemm_kernel<<<gG, blk, 0, stream>>>(adj1, node, dv1, X0);
  msg_gemm_kernel<<<gG, blk, 0, stream>>>(adj2, node, dv2, X1);
  msg_gemm_kernel<<<gG, blk, 0, stream>>>(adj3, node, dv3, X2);
  msg_gemm_kernel<<<gG, blk, 0, stream>>>(n2r, rela, nullptr, X3);
  lin_gemm_kernel<<<gG, blk, 0, stream>>>(
      X0, X1, X2, X3, Wnn + 0 * DD, Wnn + 1 * DD, Wnn + 2 * DD, Wnr + 0 * DD,
      bnn + 0 * 512, bnn + 1 * 512, bnn + 2 * 512, bnr + 0 * 512, 4, node, am,
      X4 /*node1*/);
  // -------- step 1 : rela --------
  msg_gemm_kernel<<<gG, blk, 0, stream>>>(rsub, node, nullptr, X0);
  msg_gemm_kernel<<<gG, blk, 0, stream>>>(robj, node, nullptr, X1);
  lin_gemm_kernel<<<gG, blk, 0, stream>>>(
      X0, X1, nullptr, nullptr, Wr + 0 * DD, Wr + 1 * DD, nullptr, nullptr,
      br + 0 * 512, br + 1 * 512, nullptr, nullptr, 2, rela, rm, X5 /*rela1*/);

  // -------- step 2 : node --------
  msg_gemm_kernel<<<gG, blk, 0, stream>>>(adj1, X4, dv1, X0);
  msg_gemm_kernel<<<gG, blk, 0, stream>>>(adj2, X4, dv2, X1);
  msg_gemm_kernel<<<gG, blk, 0, stream>>>(n2r, X5, nullptr, X2);
  lin_gemm_kernel<<<gG, blk, 0, stream>>>(
      X0, X1, X2, nullptr, Wnn + 3 * DD, Wnn + 4 * DD, Wnr + 1 * DD, nullptr,
      bnn + 3 * 512, bnn + 4 * 512, bnr + 1 * 512, nullptr, 3, X4, am,
      X3 /*node2*/);
  // -------- step 2 : rela --------
  msg_gemm_kernel<<<gG, blk, 0, stream>>>(rsub, X4, nullptr, X0);
  msg_gemm_kernel<<<gG, blk, 0, stream>>>(robj, X4, nullptr, X1);
  lin_gemm_kernel<<<gG, blk, 0, stream>>>(
      X0, X1, nullptr, nullptr, Wr + 2 * DD, Wr + 3 * DD, nullptr, nullptr,
      br + 2 * 512, br + 3 * 512, nullptr, nullptr, 2, X5, rm, X2 /*rela2*/);

  // -------- step 3 : node --------
  msg_gemm_kernel<<<gG, blk, 0, stream>>>(adj1, X3, dv1, X0);
  msg_gemm_kernel<<<gG, blk, 0, stream>>>(n2r, X2, nullptr, X1);
  lin_gemm_kernel<<<gG, blk, 0, stream>>>(
      X0, X1, nullptr, nullptr, Wnn + 5 * DD, Wnr + 2 * DD, nullptr, nullptr,
      bnn + 5 * 512, bnr + 2 * 512, nullptr, nullptr, 2, X3, am, outN);
  // -------- step 3 : rela --------
  msg_gemm_kernel<<<gG, blk, 0, stream>>>(rsub, X3, nullptr, X0);
  msg_gemm_kernel<<<gG, blk, 0, stream>>>(robj, X3, nullptr, X1);
  lin_gemm_kernel<<<gG, blk, 0, stream>>>(
      X0, X1, nullptr, nullptr, Wr + 4 * DD, Wr + 5 * DD, nullptr, nullptr,
      br + 4 * 512, br + 5 * 512, nullptr, nullptr, 2, X2, rm, outR);
}